// Pooling_8684423873054
// MI455X (gfx1250) — compile-verified
//
#include <hip/hip_runtime.h>
#include <hip/hip_bf16.h>

// ---------------------------------------------------------------------------
// MI455X (gfx1250) implementation.
//  - fp32 WMMA (v_wmma_f32_16x16x4_f32) for the two GEMMs: x@W_conv (50000x128x144)
//    and h@W_gather (50000x128x128). Workload is HBM/atomic bound (~1GB traffic
//    ~45us @ 23.3TB/s), GEMM flops are negligible -> keep fp32 exactness.
//  - seg_mean via global_atomic_add_f32.
//  - jnp.unique(sorted, zero-padded) reproduced with a 625Mbit bitmap,
//    popcount block sums, exclusive scan, ordered emission.
//  - bloom_pos / W_bloom are dead code in the reference and skipped.
// ---------------------------------------------------------------------------

typedef __attribute__((ext_vector_type(2))) float v2f;
typedef __attribute__((ext_vector_type(8))) float v8f;

#define N_NODES 50000
#define DEG 8
#define E_EDGES (N_NODES * DEG)
#define KDUP 2
#define C_CL 25000
#define DD_IN 128
#define DD_OUT 128
#define DD_BLOOM 16
#define DD_INTER 144
#define NP (N_NODES * KDUP)              // 100000 bloom points
#define DK (DEG * KDUP)                  // 16
static const size_t NPAIR = (size_t)N_NODES * DK * DK;     // 12,800,000
static const size_t BITWORDS = (size_t)C_CL * C_CL / 32;   // 19,531,250
#define SCAN_WPB 8192
static const int SCAN_NBLK = (int)((BITWORDS + SCAN_WPB - 1) / SCAN_WPB); // 2385

// workspace layout (4-byte word offsets)
static const size_t WS_XW     = 0;                                  // N*144
static const size_t WS_OUT    = WS_XW   + (size_t)N_NODES * DD_INTER; // N*144 sum->mean
static const size_t WS_CNT    = WS_OUT  + (size_t)N_NODES * DD_INTER; // N
static const size_t WS_HG     = WS_CNT  + (size_t)N_NODES;            // N*128
static const size_t WS_CPOS   = WS_HG   + (size_t)N_NODES * DD_OUT;   // C*3
static const size_t WS_CCNT   = WS_CPOS + (size_t)C_CL * 3;           // C
static const size_t WS_XSUM   = WS_CCNT + (size_t)C_CL;               // C*128
static const size_t WS_CB     = WS_XSUM + (size_t)C_CL * DD_OUT;      // N*16 ints
static const size_t WS_BITMAP = WS_CB   + (size_t)N_NODES * DK;       // 19,531,250
static const size_t WS_BSUM   = WS_BITMAP + BITWORDS;                 // SCAN_NBLK+1
static const size_t WS_END    = WS_BSUM + (size_t)SCAN_NBLK + 8;

// output layout (elements)
static const size_t O_XNEW   = 0;                                   // C*128
static const size_t O_NEWPOS = (size_t)C_CL * DD_OUT;               // C*3
static const size_t O_EI     = O_NEWPOS + (size_t)C_CL * 3;         // 2*NPAIR ints
static const size_t O_ATTR   = O_EI + 2 * NPAIR;                    // 3*NPAIR floats

__global__ void k_zero(float* __restrict__ p, size_t n) {
  size_t i = (size_t)blockIdx.x * blockDim.x + threadIdx.x;
  size_t stride = (size_t)gridDim.x * blockDim.x;
  for (; i < n; i += stride) p[i] = 0.0f;
}

// Out[M x P] = A[M x K] @ B[K x P], fp32 WMMA, all dims multiples of 16 (K of 4).
__global__ void k_gemm_wmma_f32(const float* __restrict__ A, const float* __restrict__ B,
                                float* __restrict__ Out,
                                int M, int Kdim, int P, int lda, int ldb, int ldc) {
  int wave = blockIdx.x * (blockDim.x >> 5) + (threadIdx.x >> 5);
  int tilesP = P >> 4;
  int numTiles = (M >> 4) * tilesP;
  if (wave >= numTiles) return;            // wave-uniform: EXEC stays all-ones
  int tm = wave / tilesP, tp = wave % tilesP;
  int lane = threadIdx.x & 31;
  int half = lane >> 4, l16 = lane & 15;
  const float* Ap = A + (size_t)(tm * 16 + l16) * lda + half * 2;
  const float* Bp = B + (size_t)(half * 2) * ldb + (tp * 16 + l16);
  v8f acc = {0.f, 0.f, 0.f, 0.f, 0.f, 0.f, 0.f, 0.f};
  for (int k = 0; k < Kdim; k += 4) {
    v2f a, b;
    a.x = Ap[k];
    a.y = Ap[k + 1];
    b.x = Bp[(size_t)k * ldb];
    b.y = Bp[(size_t)(k + 1) * ldb];
    acc = __builtin_amdgcn_wmma_f32_16x16x4_f32(false, a, false, b, (short)0, acc,
                                                false, false);
  }
  float* Cp = Out + (size_t)(tm * 16 + half * 8) * ldc + tp * 16 + l16;
#pragma unroll
  for (int j = 0; j < 8; ++j) Cp[(size_t)j * ldc] = acc[j];
}

// msg = xW[src] + edge_attr@W_cedge, scatter-add to sum[dst], count edges.
__global__ void k_msg_scatter(const float* __restrict__ xW, const float* __restrict__ edge_attr,
                              const int* __restrict__ dstIdx, const float* __restrict__ Wce,
                              float* __restrict__ sum, float* __restrict__ cnt) {
  int t = blockIdx.x * blockDim.x + threadIdx.x;
  if (t >= N_NODES * DD_INTER) return;
  int n = t / DD_INTER, c = t % DD_INTER;
  float w0 = Wce[c], w1 = Wce[DD_INTER + c], w2 = Wce[2 * DD_INTER + c];
  float base = xW[(size_t)n * DD_INTER + c];
#pragma unroll
  for (int d = 0; d < DEG; ++d) {
    int e = n * DEG + d;
    const float* ea = edge_attr + (size_t)e * 3;
    float m = base + ea[0] * w0 + ea[1] * w1 + ea[2] * w2;
    int dv = dstIdx[e];
    atomicAdd(&sum[(size_t)dv * DD_INTER + c], m);
    if (c == 0) atomicAdd(&cnt[dv], 1.0f);
  }
}

__global__ void k_mean_inplace(float* __restrict__ sum, const float* __restrict__ cnt) {
  int t = blockIdx.x * blockDim.x + threadIdx.x;
  if (t >= N_NODES * DD_INTER) return;
  sum[t] /= fmaxf(cnt[t / DD_INTER], 1.0f);
}

__global__ void k_cluster_pos(const float* __restrict__ pos, const int* __restrict__ clusters,
                              float* __restrict__ cpos, float* __restrict__ ccnt) {
  int i = blockIdx.x * blockDim.x + threadIdx.x;
  if (i >= NP) return;
  int node = i >> 1;
  int cl = clusters[i];
  atomicAdd(&cpos[(size_t)cl * 3 + 0], pos[(size_t)node * 3 + 0]);
  atomicAdd(&cpos[(size_t)cl * 3 + 1], pos[(size_t)node * 3 + 1]);
  atomicAdd(&cpos[(size_t)cl * 3 + 2], pos[(size_t)node * 3 + 2]);
  atomicAdd(&ccnt[cl], 1.0f);
}

__global__ void k_newpos(const float* __restrict__ cpos, const float* __restrict__ ccnt,
                         float* __restrict__ newpos) {
  int c = blockIdx.x * blockDim.x + threadIdx.x;
  if (c >= C_CL) return;
  float inv = 1.0f / fmaxf(ccnt[c], 1.0f);
  newpos[(size_t)c * 3 + 0] = cpos[(size_t)c * 3 + 0] * inv;
  newpos[(size_t)c * 3 + 1] = cpos[(size_t)c * 3 + 1] * inv;
  newpos[(size_t)c * 3 + 2] = cpos[(size_t)c * 3 + 2] * inv;
}

// gmsg = hG[node] + (pos[node]-new_pos[cl])@W_gedge, scatter-add to xsum[cl].
__global__ void k_gather_scatter(const float* __restrict__ pos, const int* __restrict__ clusters,
                                 const float* __restrict__ hG, const float* __restrict__ newpos,
                                 const float* __restrict__ Wge, float* __restrict__ xsum) {
  size_t t = (size_t)blockIdx.x * blockDim.x + threadIdx.x;
  if (t >= (size_t)NP * DD_OUT) return;
  int i = (int)(t >> 7), c = (int)(t & 127);
  int node = i >> 1;
  int cl = clusters[i];
  float g0 = pos[(size_t)node * 3 + 0] - newpos[(size_t)cl * 3 + 0];
  float g1 = pos[(size_t)node * 3 + 1] - newpos[(size_t)cl * 3 + 1];
  float g2 = pos[(size_t)node * 3 + 2] - newpos[(size_t)cl * 3 + 2];
  float gm = hG[(size_t)node * DD_OUT + c] + g0 * Wge[c] + g1 * Wge[DD_OUT + c] +
             g2 * Wge[2 * DD_OUT + c];
  atomicAdd(&xsum[(size_t)cl * DD_OUT + c], gm);
}

__global__ void k_xnew(const float* __restrict__ xsum, const float* __restrict__ ccnt,
                       float* __restrict__ xnew) {
  int t = blockIdx.x * blockDim.x + threadIdx.x;
  if (t >= C_CL * DD_OUT) return;
  xnew[t] = xsum[t] / fmaxf(ccnt[t >> 7], 1.0f);
}

__global__ void k_cb(const int* __restrict__ clusters, const int* __restrict__ dstIdx,
                     int* __restrict__ cb) {
  int t = blockIdx.x * blockDim.x + threadIdx.x;
  if (t >= N_NODES * DK) return;
  int n = t >> 4, q = t & 15;
  int d = q >> 1, k = q & 1;
  cb[t] = clusters[(size_t)dstIdx[n * DEG + d] * KDUP + k];
}

__global__ void k_fill_edges(int* __restrict__ ei0, int* __restrict__ ei1,
                             float* __restrict__ attr) {
  size_t t = (size_t)blockIdx.x * blockDim.x + threadIdx.x;
  if (t >= NPAIR) return;
  ei0[t] = 0;
  ei1[t] = 0;
  attr[t * 3 + 0] = 0.0f;
  attr[t * 3 + 1] = 0.0f;
  attr[t * 3 + 2] = 0.0f;
}

__global__ void k_pairs_bitmap(const int* __restrict__ cb, unsigned* __restrict__ bm) {
  size_t t = (size_t)blockIdx.x * blockDim.x + threadIdx.x;
  if (t >= NPAIR) return;
  int n = (int)(t >> 8);
  int r = (int)(t & 255);
  unsigned s = (unsigned)cb[(size_t)n * DK + (r >> 4)];
  unsigned d = (unsigned)cb[(size_t)n * DK + (r & 15)];
  unsigned enc = s * (unsigned)C_CL + d;
  atomicOr(&bm[enc >> 5], 1u << (enc & 31));
}

__global__ void k_bits_blocksum(const unsigned* __restrict__ bm, unsigned* __restrict__ bsum) {
  __shared__ unsigned sh[256];
  size_t base = (size_t)blockIdx.x * SCAN_WPB;
  unsigned s = 0;
#pragma unroll
  for (int i = 0; i < 32; ++i) {
    size_t w = base + (size_t)i * 256 + threadIdx.x;
    s += (w < BITWORDS) ? (unsigned)__popc(bm[w]) : 0u;
  }
  sh[threadIdx.x] = s;
  __syncthreads();
  for (int off = 128; off > 0; off >>= 1) {
    if (threadIdx.x < off) sh[threadIdx.x] += sh[threadIdx.x + off];
    __syncthreads();
  }
  if (threadIdx.x == 0) bsum[blockIdx.x] = sh[0];
}

__global__ void k_scan(unsigned* __restrict__ bsum, int nblk) {
  if (blockIdx.x == 0 && threadIdx.x == 0) {
    unsigned run = 0;
    for (int b = 0; b < nblk; ++b) {
      unsigned t = bsum[b];
      bsum[b] = run;
      run += t;
    }
    bsum[nblk] = run;
  }
}

// Ordered emission of sorted unique enc values + edge_index + edge_attr.
__global__ void k_emit(const unsigned* __restrict__ bm, const unsigned* __restrict__ bsum,
                       const float* __restrict__ newpos, int* __restrict__ ei0,
                       int* __restrict__ ei1, float* __restrict__ attr) {
  __shared__ unsigned sh[256];
  int tid = threadIdx.x;
  size_t wbase = (size_t)blockIdx.x * SCAN_WPB + (size_t)tid * 32;
  unsigned words[32];
  unsigned mycnt = 0;
#pragma unroll
  for (int i = 0; i < 32; ++i) {
    size_t w = wbase + i;
    unsigned v = (w < BITWORDS) ? bm[w] : 0u;
    words[i] = v;
    mycnt += (unsigned)__popc(v);
  }
  sh[tid] = mycnt;
  __syncthreads();
  for (int off = 1; off < 256; off <<= 1) {
    unsigned v = 0;
    if (tid >= off) v = sh[tid - off];
    __syncthreads();
    sh[tid] += v;
    __syncthreads();
  }
  unsigned p = bsum[blockIdx.x] + sh[tid] - mycnt;  // exclusive offset
#pragma unroll 1
  for (int i = 0; i < 32; ++i) {
    unsigned v = words[i];
    unsigned encBase = (unsigned)((wbase + (size_t)i) << 5);
    while (v) {
      int b = __ffs(v) - 1;
      v &= v - 1;
      unsigned enc = encBase + (unsigned)b;
      unsigned hi = enc / (unsigned)C_CL;
      unsigned lo = enc % (unsigned)C_CL;
      ei0[p] = (int)hi;
      ei1[p] = (int)lo;
      attr[(size_t)p * 3 + 0] = newpos[(size_t)lo * 3 + 0] - newpos[(size_t)hi * 3 + 0];
      attr[(size_t)p * 3 + 1] = newpos[(size_t)lo * 3 + 1] - newpos[(size_t)hi * 3 + 1];
      attr[(size_t)p * 3 + 2] = newpos[(size_t)lo * 3 + 2] - newpos[(size_t)hi * 3 + 2];
      ++p;
    }
  }
}

static inline int nblocks(size_t n, int bs) { return (int)((n + bs - 1) / bs); }

extern "C" void kernel_launch(void* const* d_in, const int* in_sizes, int n_in,
                              void* d_out, int out_size, void* d_ws, size_t ws_size,
                              hipStream_t stream) {
  const float* x         = (const float*)d_in[0];
  const float* pos       = (const float*)d_in[1];
  const int*   edge_idx  = (const int*)d_in[2];
  const float* edge_attr = (const float*)d_in[3];
  // d_in[4] = batch (unused), d_in[8] = W_bloom (dead code in reference)
  const int*   clusters  = (const int*)d_in[5];
  const float* W_conv    = (const float*)d_in[6];
  const float* W_cedge   = (const float*)d_in[7];
  const float* W_gather  = (const float*)d_in[9];
  const float* W_gedge   = (const float*)d_in[10];
  const int* dst = edge_idx + E_EDGES;

  float* ws = (float*)d_ws;
  float* out_f = (float*)d_out;
  int* out_i = (int*)d_out;

  float* xW     = ws + WS_XW;
  float* outbuf = ws + WS_OUT;   // sum -> mean (sph|h), h at col 16
  float* cnt    = ws + WS_CNT;
  float* hG     = ws + WS_HG;
  float* cpos   = ws + WS_CPOS;
  float* ccnt   = ws + WS_CCNT;
  float* xsum   = ws + WS_XSUM;
  int* cb       = (int*)(ws + WS_CB);
  unsigned* bm  = (unsigned*)(ws + WS_BITMAP);
  unsigned* bsum = (unsigned*)(ws + WS_BSUM);

  float* newpos = out_f + O_NEWPOS;
  int* ei0 = out_i + O_EI;
  int* ei1 = out_i + O_EI + NPAIR;
  float* attr = out_f + O_ATTR;

  // 0) zero accumulators + bitmap + scan buffer
  size_t zn = WS_END - WS_OUT;
  k_zero<<<4096, 256, 0, stream>>>(ws + WS_OUT, zn);

  // 1) xW = x @ W_conv   (50000x128 @ 128x144), fp32 WMMA
  {
    int tiles = (N_NODES / 16) * (DD_INTER / 16);  // 3125*9
    k_gemm_wmma_f32<<<nblocks(tiles, 8), 256, 0, stream>>>(
        x, W_conv, xW, N_NODES, DD_IN, DD_INTER, DD_IN, DD_INTER, DD_INTER);
  }

  // 2) scatter messages to dst, count edges
  k_msg_scatter<<<nblocks((size_t)N_NODES * DD_INTER, 256), 256, 0, stream>>>(
      xW, edge_attr, dst, W_cedge, outbuf, cnt);

  // 3) mean
  k_mean_inplace<<<nblocks((size_t)N_NODES * DD_INTER, 256), 256, 0, stream>>>(outbuf, cnt);

  // 4) cluster position means -> new_pos (written straight to d_out)
  k_cluster_pos<<<nblocks(NP, 256), 256, 0, stream>>>(pos, clusters, cpos, ccnt);
  k_newpos<<<nblocks(C_CL, 256), 256, 0, stream>>>(cpos, ccnt, newpos);

  // 5) hG = h @ W_gather  (h = outbuf cols 16..143, lda=144), fp32 WMMA
  {
    int tiles = (N_NODES / 16) * (DD_OUT / 16);  // 3125*8
    k_gemm_wmma_f32<<<nblocks(tiles, 8), 256, 0, stream>>>(
        outbuf + DD_BLOOM, W_gather, hG, N_NODES, DD_OUT, DD_OUT, DD_INTER, DD_OUT, DD_OUT);
  }

  // 6) gather messages -> x_new (cluster counts identical to step 4)
  k_gather_scatter<<<nblocks((size_t)NP * DD_OUT, 256), 256, 0, stream>>>(
      pos, clusters, hG, newpos, W_gedge, xsum);
  k_xnew<<<nblocks((size_t)C_CL * DD_OUT, 256), 256, 0, stream>>>(xsum, ccnt, out_f + O_XNEW);

  // 7) edge reconstruction: cb, pair bitmap, sorted-unique emission
  k_cb<<<nblocks((size_t)N_NODES * DK, 256), 256, 0, stream>>>(clusters, dst, cb);
  k_fill_edges<<<nblocks(NPAIR, 256), 256, 0, stream>>>(ei0, ei1, attr);
  k_pairs_bitmap<<<nblocks(NPAIR, 256), 256, 0, stream>>>(cb, bm);
  k_bits_blocksum<<<SCAN_NBLK, 256, 0, stream>>>(bm, bsum);
  k_scan<<<1, 32, 0, stream>>>(bsum, SCAN_NBLK);
  k_emit<<<SCAN_NBLK, 256, 0, stream>>>(bm, bsum, newpos, ei0, ei1, attr);
}